// SparseLinear_41197326303441
// MI455X (gfx1250) — compile-verified
//
#include <hip/hip_runtime.h>

typedef __attribute__((ext_vector_type(2))) float v2f;
typedef __attribute__((ext_vector_type(8))) float v8f;

#define G1C 512     // graph_size1 (destination block rows)
#define EC  16384   // edges
#define S1C 8192    // input features
#define S2C 8192    // output features
#define BC  256     // batch
#define CHUNK 8     // edges staged into LDS per barrier round

// ---------------- bucketing passes (tiny: 16K edges) ----------------

__global__ void sl_init_counts(int* __restrict__ counts) {
    int i = blockIdx.x * blockDim.x + threadIdx.x;
    if (i < G1C) counts[i] = 0;
}

__global__ void sl_count(const int* __restrict__ e0, int* __restrict__ counts) {
    int e = blockIdx.x * blockDim.x + threadIdx.x;
    if (e < EC) atomicAdd(&counts[e0[e]], 1);
}

__global__ void sl_scan(const int* __restrict__ counts,
                        int* __restrict__ offsets,
                        int* __restrict__ cursor) {
    if (blockIdx.x == 0 && threadIdx.x == 0) {
        int acc = 0;
        for (int i = 0; i < G1C; ++i) {
            offsets[i] = acc;
            cursor[i]  = acc;
            acc += counts[i];
        }
        offsets[G1C] = acc;
    }
}

__global__ void sl_scatter(const int* __restrict__ e0,
                           int* __restrict__ cursor,
                           int* __restrict__ bucket) {
    int e = blockIdx.x * blockDim.x + threadIdx.x;
    if (e < EC) {
        int pos = atomicAdd(&cursor[e0[e]], 1);
        bucket[pos] = e;
    }
}

// ---------------- main block-SpMM via V_WMMA_F32_16X16X4_F32 ----------------
//
// One workgroup per t0, covering the full batch (256 rows). 8 waves, each
// wave owns TWO 16x16 output tiles (32 batch rows x 16 features), giving two
// independent WMMA accumulator chains and 2x B-operand reuse from LDS.
// V blocks are staged CHUNK edges at a time (8 KB LDS) to amortize barriers.
// values: streamed exactly once -> non-temporal. out: write-once -> NT store.

__global__ __launch_bounds__(256)
void sl_spmm(const float* __restrict__ x,
             const float* __restrict__ values,
             const float* __restrict__ bias,
             const int* __restrict__ e1,
             const int* __restrict__ offsets,
             const int* __restrict__ bucket,
             float* __restrict__ out) {
    __shared__ float Vt[CHUNK * 256];  // CHUNK 16x16 f32 blocks, i-major

    const int t0   = blockIdx.x;
    const int tid  = threadIdx.x;
    const int lane = tid & 31;
    const int wave = tid >> 5;

    const int batchBase = wave * 32;     // 32 batch rows per wave (2 tiles)
    const int m  = lane & 15;            // A-matrix row (batch) within tile
    const int n  = lane & 15;            // B/C/D column (feature j)
    const int kb = (lane >> 4) << 1;     // K sub-pair: lanes 0-15 -> {0,1}, 16-31 -> {2,3}

    // Seed accumulators with bias (added exactly once per output element).
    v8f acc0, acc1;
    const float bv = bias[t0 * 16 + n];
#pragma unroll
    for (int p = 0; p < 8; ++p) { acc0[p] = bv; acc1[p] = bv; }

    const int beg = offsets[t0];
    const int end = offsets[t0 + 1];

    const float* xrow0 = x + (size_t)(batchBase + m) * S1C;        // tile0 rows
    const float* xrow1 = xrow0 + (size_t)16 * S1C;                 // tile1 rows

    for (int base = beg; base < end; base += CHUNK) {
        const int cnt = min(CHUNK, end - base);

        __syncthreads();  // previous round's LDS readers done
        for (int c = 0; c < cnt; ++c) {
            const int e = bucket[base + c];   // workgroup-uniform -> scalar load
            Vt[c * 256 + tid] =
                __builtin_nontemporal_load(values + (size_t)e * 256 + tid);
        }
        __syncthreads();

        for (int c = 0; c < cnt; ++c) {
            const int e  = bucket[base + c];
            const int t1 = e1[e];

            const float* xt0 = xrow0 + t1 * 16 + kb;
            const float* xt1 = xrow1 + t1 * 16 + kb;
            const float* vb  = &Vt[c * 256];

#pragma unroll
            for (int k0 = 0; k0 < 16; k0 += 4) {
                // A 16x4: lane l holds X[m, k0+kb], X[m, k0+kb+1] (8B aligned)
                v2f a0 = *(const v2f*)(xt0 + k0);
                v2f a1 = *(const v2f*)(xt1 + k0);
                // B 4x16: lane l holds V[k0+kb, n], V[k0+kb+1, n] (shared by both tiles)
                v2f bm;
                bm[0] = vb[(k0 + kb) * 16 + n];
                bm[1] = vb[(k0 + kb + 1) * 16 + n];
                acc0 = __builtin_amdgcn_wmma_f32_16x16x4_f32(
                    false, a0, false, bm, (short)0, acc0, false, false);
                acc1 = __builtin_amdgcn_wmma_f32_16x16x4_f32(
                    false, a1, false, bm, (short)0, acc1, false, false);
            }
        }
    }

    // D layout: VGPR p -> M = p (lanes 0-15) / p+8 (lanes 16-31), N = lane&15.
    const int row0 = batchBase + ((lane >> 4) << 3);
    float* outp0 = out + (size_t)row0 * S2C + t0 * 16 + n;
    float* outp1 = outp0 + (size_t)16 * S2C;
#pragma unroll
    for (int p = 0; p < 8; ++p) {
        __builtin_nontemporal_store(acc0[p], outp0 + (size_t)p * S2C);
        __builtin_nontemporal_store(acc1[p], outp1 + (size_t)p * S2C);
    }
}

// ---------------- launcher ----------------

extern "C" void kernel_launch(void* const* d_in, const int* in_sizes, int n_in,
                              void* d_out, int out_size, void* d_ws, size_t ws_size,
                              hipStream_t stream) {
    const float* x      = (const float*)d_in[0];
    const float* values = (const float*)d_in[1];
    const float* bias   = (const float*)d_in[2];
    const int*   edge   = (const int*)d_in[3];
    const int*   e0     = edge;        // edge_index[0] : destination block rows
    const int*   e1     = edge + EC;   // edge_index[1] : source block cols
    float*       out    = (float*)d_out;

    // workspace layout (ints): counts[512] | offsets[513] | cursor[512] | bucket[16384]
    int* counts  = (int*)d_ws;
    int* offsets = counts + G1C;
    int* cursor  = offsets + (G1C + 1);
    int* bucket  = cursor + G1C;

    sl_init_counts<<<(G1C + 255) / 256, 256, 0, stream>>>(counts);
    sl_count<<<EC / 256, 256, 0, stream>>>(e0, counts);
    sl_scan<<<1, 1, 0, stream>>>(counts, offsets, cursor);
    sl_scatter<<<EC / 256, 256, 0, stream>>>(e0, cursor, bucket);
    sl_spmm<<<G1C, 256, 0, stream>>>(x, values, bias, e1, offsets, bucket, out);
}